// GCN2_1357209666150
// MI455X (gfx1250) — compile-verified
//
#include <hip/hip_runtime.h>
#include <math.h>
#include <stdint.h>

#define NNODES  100000
#define EDGES   1000000
#define INFEAT  128
#define HIDF    64
#define NLAYER  4
#define ALPHA_  0.1f
#define BN_EPS_ 1e-5f
#define NTILES  (NNODES / 16)   // 6250, exact

typedef float v2f __attribute__((ext_vector_type(2)));
typedef float v8f __attribute__((ext_vector_type(8)));

__device__ __forceinline__ v8f wmma4(v2f a, v2f b, v8f c) {
    // V_WMMA_F32_16X16X4_F32 : D = A(16x4 f32) * B(4x16 f32) + C(16x16 f32)
    return __builtin_amdgcn_wmma_f32_16x16x4_f32(
        /*neg_a=*/false, a, /*neg_b=*/false, b,
        /*c_mod=*/(short)0, c, /*reuse_a=*/false, /*reuse_b=*/false);
}

// CDNA5 async global->LDS copy: 16B per lane, no VGPR round trip, ASYNCcnt-tracked.
__device__ __forceinline__ void async_copy_b128(uint32_t lds_off, const void* gaddr) {
    asm volatile("global_load_async_to_lds_b128 %0, %1, off"
                 :: "v"(lds_off), "v"((unsigned long long)(uintptr_t)gaddr)
                 : "memory");
}

__device__ __forceinline__ void wait_async0() {
#if __has_builtin(__builtin_amdgcn_s_wait_asynccnt)
    __builtin_amdgcn_s_wait_asynccnt(0);
#else
    asm volatile("s_wait_asynccnt 0x0" ::: "memory");
#endif
}

// ---------------------------------------------------------------- utilities
__global__ void fill_kernel(float* __restrict__ p, int n, float v) {
    int i = blockIdx.x * blockDim.x + threadIdx.x;
    if (i < n) p[i] = v;
}

__global__ void degree_kernel(const int* __restrict__ row, float* __restrict__ deg, int ne) {
    int e = blockIdx.x * blockDim.x + threadIdx.x;
    if (e < ne) atomicAdd(&deg[row[e]], 1.0f);   // exact: integer-valued fp32 sums
}

__global__ void dinv_kernel(const float* __restrict__ deg, float* __restrict__ dinv, int n) {
    int i = blockIdx.x * blockDim.x + threadIdx.x;
    if (i < n) {
        float d = deg[i];
        dinv[i] = (d > 0.0f) ? rsqrtf(d) : 0.0f;
    }
}

// per-edge weight D^-1/2 A D^-1/2, layer-invariant -> compute once
__global__ void ew_kernel(const int* __restrict__ row, const int* __restrict__ col,
                          const float* __restrict__ dinv, float* __restrict__ ew, int ne) {
    int e = blockIdx.x * blockDim.x + threadIdx.x;
    if (e < ne) ew[e] = dinv[row[e]] * dinv[col[e]];
}

// ------------------------------------------------- GEMM0: relu(x@W0 + b0)
// one wave per 16-row tile; A tile (16x128) staged in LDS via async copies.
__global__ __launch_bounds__(128)
void gemm0_kernel(const float* __restrict__ x, const float* __restrict__ W0,
                  const float* __restrict__ b0, float* __restrict__ xcur,
                  float* __restrict__ x0out) {
    __shared__ float sA[4][16 * INFEAT];      // 4 waves * 8KB = 32KB
    const int tid  = threadIdx.x;
    const int wave = tid >> 5;
    const int lane = tid & 31;
    const int tile = blockIdx.x * 4 + wave;
    if (tile >= NTILES) return;               // whole-wave guard; no block barrier used
    const int rbase = tile * 16;

    {   // stage 16x128 tile: 512 float4 -> 16 async b128 per lane-slot
        uint32_t lbase = (uint32_t)(uintptr_t)&sA[wave][0];
        const char* g = (const char*)(x + (size_t)rbase * INFEAT);
        #pragma unroll
        for (int i = 0; i < 16; i++) {
            int idx = i * 32 + lane;
            async_copy_b128(lbase + (uint32_t)idx * 16u, g + (size_t)idx * 16);
        }
        wait_async0();                        // wave-local: ASYNCcnt==0 => LDS ready
    }

    const int half = lane >> 4;               // 0: K={0,1}, 1: K={2,3}
    const int m    = lane & 15;
    v8f acc[4];
    #pragma unroll
    for (int nt = 0; nt < 4; nt++)
        #pragma unroll
        for (int v = 0; v < 8; v++) acc[nt][v] = 0.0f;

    for (int kk = 0; kk < INFEAT / 4; kk++) {
        const int kb = kk * 4 + half * 2;
        const float* ap = &sA[wave][m * INFEAT + kb];
        v2f a; a[0] = ap[0]; a[1] = ap[1];
        #pragma unroll
        for (int nt = 0; nt < 4; nt++) {
            const float* bp = &W0[kb * HIDF + nt * 16 + m];
            v2f b; b[0] = bp[0]; b[1] = bp[HIDF];
            acc[nt] = wmma4(a, b, acc[nt]);
        }
    }
    #pragma unroll
    for (int nt = 0; nt < 4; nt++) {
        const int c = nt * 16 + m;
        const float bias = b0[c];
        #pragma unroll
        for (int v = 0; v < 8; v++) {
            const int r = rbase + v + half * 8;
            float val = acc[nt][v] + bias;
            val = val > 0.0f ? val : 0.0f;
            xcur[(size_t)r * HIDF + c]  = val;
            x0out[(size_t)r * HIDF + c] = val;
        }
    }
}

// ------------------------------------------------- SpMM: agg[row] += ew * x[col]
// one wave (32 lanes x float2) per edge: edge metadata coalesces to one request/wave;
// agg + x both L2-resident (25.6 MB each vs 192 MB L2).
__global__ void spmm_kernel(const int* __restrict__ row, const int* __restrict__ col,
                            const float* __restrict__ ew, const float* __restrict__ xin,
                            float* __restrict__ agg) {
    int t = blockIdx.x * blockDim.x + threadIdx.x;   // < 32M, fits int
    int e = t >> 5;
    int j = t & 31;
    if (e < EDGES) {
        int r = row[e], c = col[e];
        float w = ew[e];
        const float2* xp = (const float2*)(xin + (size_t)c * HIDF);
        float2 v = xp[j];
        float* ap = agg + (size_t)r * HIDF + j * 2;
        atomicAdd(ap,     w * v.x);
        atomicAdd(ap + 1, w * v.y);
    }
}

// ------------------------------------------------- fused layer dense:
//   s = (1-a)*agg + a*x0   (computed into LDS)
//   h = (1-beta)*s + beta*(s @ W)   -> written back over agg
//   BN partial sums reduced via LDS, one global atomic per feature per block
__global__ __launch_bounds__(128)
void layer_dense_kernel(const float* __restrict__ x0, float* __restrict__ agg,
                        const float* __restrict__ W, float beta,
                        float* __restrict__ gsum, float* __restrict__ gsumsq) {
    __shared__ float sS[4][16 * HIDF];        // 4 waves * 4KB
    __shared__ float bsum[HIDF], bsumsq[HIDF];
    const int tid  = threadIdx.x;
    const int wave = tid >> 5;
    const int lane = tid & 31;
    const int tile = blockIdx.x * 4 + wave;
    const bool valid = (tile < NTILES);
    const int rbase = tile * 16;

    if (tid < 64) bsum[tid] = 0.0f;
    else          bsumsq[tid - 64] = 0.0f;

    if (valid) {
        const float4* ag = (const float4*)(agg + (size_t)rbase * HIDF);
        const float4* xp = (const float4*)(x0  + (size_t)rbase * HIDF);
        float4* dst = (float4*)sS[wave];
        for (int i = lane; i < 16 * HIDF / 4; i += 32) {
            float4 a = ag[i], b = xp[i], s;
            s.x = (1.0f - ALPHA_) * a.x + ALPHA_ * b.x;
            s.y = (1.0f - ALPHA_) * a.y + ALPHA_ * b.y;
            s.z = (1.0f - ALPHA_) * a.z + ALPHA_ * b.z;
            s.w = (1.0f - ALPHA_) * a.w + ALPHA_ * b.w;
            dst[i] = s;
        }
    }
    __syncthreads();

    if (valid) {
        const int half = lane >> 4;
        const int m    = lane & 15;
        v8f acc[4];
        #pragma unroll
        for (int nt = 0; nt < 4; nt++)
            #pragma unroll
            for (int v = 0; v < 8; v++) acc[nt][v] = 0.0f;

        for (int kk = 0; kk < HIDF / 4; kk++) {
            const int kb = kk * 4 + half * 2;
            const float* ap = &sS[wave][m * HIDF + kb];
            v2f a; a[0] = ap[0]; a[1] = ap[1];
            #pragma unroll
            for (int nt = 0; nt < 4; nt++) {
                const float* bp = &W[kb * HIDF + nt * 16 + m];
                v2f b; b[0] = bp[0]; b[1] = bp[HIDF];
                acc[nt] = wmma4(a, b, acc[nt]);
            }
        }
        const float ombeta = 1.0f - beta;
        #pragma unroll
        for (int nt = 0; nt < 4; nt++) {
            const int c = nt * 16 + m;
            float ps = 0.0f, pss = 0.0f;
            #pragma unroll
            for (int v = 0; v < 8; v++) {
                const int rr = v + half * 8;
                const int r  = rbase + rr;
                float sval = sS[wave][rr * HIDF + c];
                float h = ombeta * sval + beta * acc[nt][v];
                agg[(size_t)r * HIDF + c] = h;   // h overwrites agg
                ps  += h;
                pss += h * h;
            }
            atomicAdd(&bsum[c],  ps);
            atomicAdd(&bsumsq[c], pss);
        }
    }
    __syncthreads();
    if (tid < 64)       atomicAdd(&gsum[tid],        bsum[tid]);
    else                atomicAdd(&gsumsq[tid - 64], bsumsq[tid - 64]);
}

// ------------------------------------------------- BN + residual + ReLU (in place on xcur)
__global__ void bn_res_relu_kernel(const float* __restrict__ h, const float* __restrict__ gsum,
                                   const float* __restrict__ gsumsq,
                                   const float* __restrict__ gamma, const float* __restrict__ betab,
                                   float* __restrict__ xcur) {
    int i = blockIdx.x * blockDim.x + threadIdx.x;
    if (i < NNODES * HIDF) {
        int f = i & 63;
        const float inv_n = 1.0f / (float)NNODES;
        float mu  = gsum[f] * inv_n;
        float var = gsumsq[f] * inv_n - mu * mu;
        float sc  = rsqrtf(var + BN_EPS_) * gamma[f];
        float val = (h[i] - mu) * sc + betab[f] + xcur[i];
        xcur[i] = val > 0.0f ? val : 0.0f;
    }
}

// ------------------------------------------------- final GEMM: x@W1 + b1 -> out
__global__ __launch_bounds__(128)
void gemm_out_kernel(const float* __restrict__ x, const float* __restrict__ W1,
                     const float* __restrict__ b1, float* __restrict__ out) {
    __shared__ float sA[4][16 * HIDF];
    const int tid  = threadIdx.x;
    const int wave = tid >> 5;
    const int lane = tid & 31;
    const int tile = blockIdx.x * 4 + wave;
    if (tile >= NTILES) return;               // whole-wave guard; no block barrier used
    const int rbase = tile * 16;

    {   // stage 16x64 tile: 256 float4 -> 8 async b128 per lane-slot
        uint32_t lbase = (uint32_t)(uintptr_t)&sA[wave][0];
        const char* g = (const char*)(x + (size_t)rbase * HIDF);
        #pragma unroll
        for (int i = 0; i < 8; i++) {
            int idx = i * 32 + lane;
            async_copy_b128(lbase + (uint32_t)idx * 16u, g + (size_t)idx * 16);
        }
        wait_async0();
    }

    const int half = lane >> 4;
    const int m    = lane & 15;
    v8f acc[4];
    #pragma unroll
    for (int nt = 0; nt < 4; nt++)
        #pragma unroll
        for (int v = 0; v < 8; v++) acc[nt][v] = 0.0f;

    for (int kk = 0; kk < HIDF / 4; kk++) {
        const int kb = kk * 4 + half * 2;
        const float* ap = &sA[wave][m * HIDF + kb];
        v2f a; a[0] = ap[0]; a[1] = ap[1];
        #pragma unroll
        for (int nt = 0; nt < 4; nt++) {
            const float* bp = &W1[kb * HIDF + nt * 16 + m];
            v2f b; b[0] = bp[0]; b[1] = bp[HIDF];
            acc[nt] = wmma4(a, b, acc[nt]);
        }
    }
    #pragma unroll
    for (int nt = 0; nt < 4; nt++) {
        const int c = nt * 16 + m;
        const float bias = b1[c];
        #pragma unroll
        for (int v = 0; v < 8; v++) {
            const int r = rbase + v + half * 8;
            out[(size_t)r * HIDF + c] = acc[nt][v] + bias;
        }
    }
}

// ----------------------------------------------------------------- launcher
extern "C" void kernel_launch(void* const* d_in, const int* in_sizes, int n_in,
                              void* d_out, int out_size, void* d_ws, size_t ws_size,
                              hipStream_t stream) {
    (void)in_sizes; (void)n_in; (void)out_size; (void)ws_size;
    const float* x     = (const float*)d_in[0];
    const int*   ei    = (const int*)  d_in[1];
    const float* W0    = (const float*)d_in[2];
    const float* b0    = (const float*)d_in[3];
    const float* convw = (const float*)d_in[4];
    const float* gam   = (const float*)d_in[5];
    const float* bet   = (const float*)d_in[6];
    const float* W1    = (const float*)d_in[7];
    const float* b1    = (const float*)d_in[8];
    float* out = (float*)d_out;

    // workspace (floats): dinv[N] | xcur[N*64] | x0[N*64] | agg[N*64] | gsum[64] | gsumsq[64] | ew[E]
    float* ws     = (float*)d_ws;
    float* dinv   = ws;
    float* xcur   = dinv + NNODES;
    float* x0     = xcur + (size_t)NNODES * HIDF;
    float* agg    = x0   + (size_t)NNODES * HIDF;
    float* gsum   = agg  + (size_t)NNODES * HIDF;
    float* gsumsq = gsum + HIDF;
    float* ew     = gsumsq + HIDF;

    const int* rowp = ei;
    const int* colp = ei + EDGES;

    // degrees (reuse agg's first N floats as deg scratch) -> dinv -> per-edge weights
    fill_kernel<<<(NNODES + 255) / 256, 256, 0, stream>>>(agg, NNODES, 0.0f);
    degree_kernel<<<(EDGES + 255) / 256, 256, 0, stream>>>(rowp, agg, EDGES);
    dinv_kernel<<<(NNODES + 255) / 256, 256, 0, stream>>>(agg, dinv, NNODES);
    ew_kernel<<<(EDGES + 255) / 256, 256, 0, stream>>>(rowp, colp, dinv, ew, EDGES);

    const int gblocks = (NTILES + 3) / 4;   // 4 waves (tiles) per 128-thread block
    gemm0_kernel<<<gblocks, 128, 0, stream>>>(x, W0, b0, xcur, x0);

    for (int l = 0; l < NLAYER; l++) {
        // zero agg + gsum + gsumsq (contiguous)
        fill_kernel<<<(NNODES * HIDF + 128 + 255) / 256, 256, 0, stream>>>(
            agg, NNODES * HIDF + 128, 0.0f);
        spmm_kernel<<<(EDGES * 32) / 256, 256, 0, stream>>>(rowp, colp, ew, xcur, agg);
        float beta = logf(0.5f / (float)(l + 1) + 1.0f);
        layer_dense_kernel<<<gblocks, 128, 0, stream>>>(
            x0, agg, convw + (size_t)l * HIDF * HIDF, beta, gsum, gsumsq);
        bn_res_relu_kernel<<<(NNODES * HIDF + 255) / 256, 256, 0, stream>>>(
            agg, gsum, gsumsq, gam + l * HIDF, bet + l * HIDF, xcur);
    }

    gemm_out_kernel<<<gblocks, 128, 0, stream>>>(xcur, W1, b1, out);
}